// EtcSequenceTagger_24223615549770
// MI455X (gfx1250) — compile-verified
//
#include <hip/hip_runtime.h>

// ---- problem constants (from reference) ----
#define B_ 16
#define L_ 4096
#define G_ 256
#define R_ 84
#define W_ 169            // 2R+1
#define DMAX_ 12
#define VOCAB_ 25         // 2*D+1

#define N_L2L (B_*L_*W_)          // 11,075,584
#define N_G2G (B_*G_*G_)          //  1,048,576
#define N_L2G (B_*L_*G_)          // 16,777,216
#define N_G2L (B_*G_*L_)          // 16,777,216

// output offsets (in floats), reference tuple return order
#define OFF_L2L_MASK 0
#define OFF_G2G_MASK (OFF_L2L_MASK + N_L2L)
#define OFF_L2G_MASK (OFF_G2G_MASK + N_G2G)
#define OFF_G2L_MASK (OFF_L2G_MASK + N_L2G)
#define OFF_L2L_REL  (OFF_G2L_MASK + N_G2L)
#define OFF_G2G_REL  (OFF_L2L_REL  + N_L2L)
#define OFF_L2G_REL  (OFF_G2G_REL  + N_G2G)
#define OFF_G2L_REL  (OFF_L2G_REL  + N_L2G)

typedef __attribute__((ext_vector_type(2))) float v2f;
typedef __attribute__((ext_vector_type(4))) float v4f;
typedef __attribute__((ext_vector_type(8))) float v8f;

// ------------------------------------------------------------------
// Kernel 1: per-batch reverse cumsums (suffix sums) into workspace.
// One 1024-thread block per batch; 4 elems/thread for the L=4096 row,
// then a 256-wide scan for the G=256 row. LDS Hillis-Steele scan.
// ------------------------------------------------------------------
__global__ __launch_bounds__(1024) void seg_kernel(const int* __restrict__ lbp,
                                                   const int* __restrict__ gbp,
                                                   int* __restrict__ lseg,
                                                   int* __restrict__ gseg) {
    __shared__ int sm[1024];
    const int b = blockIdx.x;
    const int t = threadIdx.x;

    // ---- long row: suffix(i) = total - exclusive_prefix(i) ----
    int4 v = ((const int4*)(lbp + b * L_))[t];
    int s0 = v.x, s1 = s0 + v.y, s2 = s1 + v.z, s3 = s2 + v.w;
    sm[t] = s3;
    __syncthreads();
    for (int off = 1; off < 1024; off <<= 1) {
        int x = (t >= off) ? sm[t - off] : 0;
        __syncthreads();
        sm[t] += x;
        __syncthreads();
    }
    int total = sm[1023];
    int excl  = sm[t] - s3;      // exclusive prefix of this thread's 4-chunk
    int4 o;
    o.x = total - excl;
    o.y = total - (excl + s0);
    o.z = total - (excl + s1);
    o.w = total - (excl + s2);
    ((int4*)(lseg + b * L_))[t] = o;
    __syncthreads();

    // ---- glob row (256 elems) ----
    int gv = 0;
    if (t < G_) { gv = gbp[b * G_ + t]; sm[t] = gv; }
    __syncthreads();
    for (int off = 1; off < G_; off <<= 1) {
        int x = (t >= off && t < G_) ? sm[t - off] : 0;
        __syncthreads();
        if (t < G_) sm[t] += x;
        __syncthreads();
    }
    if (t < G_) {
        int gtotal = sm[G_ - 1];
        gseg[b * G_ + t] = gtotal - (sm[t] - gv);
    }
}

// ------------------------------------------------------------------
// Kernel 2: l2l mask (B,L,169) + l2l rel (B,L,169).
// Flat mapping, 4 consecutive elements per thread, NT float4 stores.
// ------------------------------------------------------------------
__global__ __launch_bounds__(256) void l2l_kernel(const int* __restrict__ lseg,
                                                  float* __restrict__ out) {
    unsigned f = (blockIdx.x * 256u + threadIdx.x) * 4u;
    unsigned k = f % 169u;
    unsigned r = f / 169u;
    int l = (int)(r & 4095u);
    int b = (int)(r >> 12);
    int seg = lseg[b * L_ + l];
    v4f mv, rv;
#pragma unroll
    for (int j = 0; j < 4; ++j) {
        int p  = l + (int)k - R_;
        int pc = p < 0 ? 0 : (p >= L_ ? L_ - 1 : p);      // clamp so load is unconditional
        int m  = (p == pc) && (lseg[b * L_ + pc] == seg);
        int rel = (int)k - R_;
        rel = rel < -DMAX_ ? -DMAX_ : (rel > DMAX_ ? DMAX_ : rel);
        int rid = rel >= 0 ? rel : (DMAX_ - rel);
        mv[j] = (float)m;
        rv[j] = (float)rid;
        if (j < 3) {
            if (++k == 169u) {
                k = 0u;
                if (++l == L_) { l = 0; ++b; }
                seg = lseg[b * L_ + l];
            }
        }
    }
    __builtin_nontemporal_store(mv, (v4f*)(out + OFF_L2L_MASK + f));
    __builtin_nontemporal_store(rv, (v4f*)(out + OFF_L2L_REL  + f));
}

// ------------------------------------------------------------------
// Kernel 3: g2g mask via V_WMMA_F32_16X16X4_F32 + g2g rel (VALU).
// eq(ti,tj) = ti*tj + (1-ti)(1-tj): A(16x4) = [t, 1-t, 0, 0],
// B(4x16) likewise. One wave per 16x16 tile; 8 waves/block; 4096 tiles.
// A layout (ISA 7.12.2): lanes0-15 hold K=0 (VGPR0) / K=1 (VGPR1),
// lanes16-31 hold K=2/K=3 (we put zeros there).
// ------------------------------------------------------------------
__global__ __launch_bounds__(256) void g2g_kernel(const int* __restrict__ gseg,
                                                  float* __restrict__ out) {
    int wid  = blockIdx.x * 8 + (threadIdx.x >> 5);   // global wave id, 0..4095
    int lane = threadIdx.x & 31;
    int b  = wid >> 8;
    int t  = wid & 255;
    int ti = t >> 4, tj = t & 15;
    int n  = lane & 15;
    int hi = lane >> 4;                               // 0: K=0/1 rows, 1: K=2/3 (zero)

    float tiv = (gseg[b * G_ + ti * 16 + n] > 0) ? 1.0f : 0.0f;
    int   segj = gseg[b * G_ + tj * 16 + n];
    float tjv = (segj > 0) ? 1.0f : 0.0f;

    v2f a, bb;
    a[0]  = hi ? 0.0f : tiv;  a[1]  = hi ? 0.0f : (1.0f - tiv);
    bb[0] = hi ? 0.0f : tjv;  bb[1] = hi ? 0.0f : (1.0f - tjv);
    v8f c = {};
    // D = A x B + 0  -> 16x16 equality tile in f32 accumulators
    c = __builtin_amdgcn_wmma_f32_16x16x4_f32(false, a, false, bb, (short)0, c,
                                              false, false);
#pragma unroll
    for (int vv = 0; vv < 8; ++vv) {
        int m  = vv + 8 * hi;                         // C/D layout: row = v + 8*half
        int gi = ti * 16 + m;
        int gj = tj * 16 + n;
        unsigned o = (unsigned)((b * G_ + gi) * G_ + gj);
        __builtin_nontemporal_store(c[vv], out + OFF_G2G_MASK + o);

        int segi = gseg[b * G_ + gi];
        int rel  = gj - gi;
        rel = rel < -DMAX_ ? -DMAX_ : (rel > DMAX_ ? DMAX_ : rel);
        int rid = rel >= 0 ? rel : (DMAX_ - rel);
        float rvf = (segi == segj) ? (float)rid : (float)(VOCAB_ + 2);
        __builtin_nontemporal_store(rvf, out + OFF_G2G_REL + o);
    }
}

// ------------------------------------------------------------------
// Kernel 4: l2g mask (B,L,G) + l2g rel. 4 consecutive g per thread.
// ------------------------------------------------------------------
__global__ __launch_bounds__(256) void l2g_kernel(const int* __restrict__ lseg,
                                                  const int* __restrict__ gseg,
                                                  const int* __restrict__ lpid,
                                                  float* __restrict__ out) {
    unsigned f = (blockIdx.x * 256u + threadIdx.x) * 4u;
    int g = (int)(f & 255u);
    unsigned r = f >> 8;
    int l = (int)(r & 4095u);
    int b = (int)(r >> 12);
    int pid = lpid[b * L_ + l];
    int lt  = lseg[b * L_ + l] > 0;
    int4 gs = *(const int4*)(gseg + b * G_ + g);
    int gsv[4] = {gs.x, gs.y, gs.z, gs.w};
    v4f mv, rv;
#pragma unroll
    for (int j = 0; j < 4; ++j) {
        int gg = g + j;
        int gt = gsv[j] > 0;
        mv[j] = (float)((lt == gt) && (pid == gg));
        rv[j] = (float)(VOCAB_ + (pid == gg ? 1 : 0));
    }
    __builtin_nontemporal_store(mv, (v4f*)(out + OFF_L2G_MASK + f));
    __builtin_nontemporal_store(rv, (v4f*)(out + OFF_L2G_REL  + f));
}

// ------------------------------------------------------------------
// Kernel 5: g2l mask (B,G,L) + g2l rel. 4 consecutive l per thread.
// ------------------------------------------------------------------
__global__ __launch_bounds__(256) void g2l_kernel(const int* __restrict__ lseg,
                                                  const int* __restrict__ gseg,
                                                  const int* __restrict__ lpid,
                                                  float* __restrict__ out) {
    unsigned f = (blockIdx.x * 256u + threadIdx.x) * 4u;
    int l = (int)(f & 4095u);
    unsigned r = f >> 12;
    int g = (int)(r & 255u);
    int b = (int)(r >> 8);
    int gt = gseg[b * G_ + g] > 0;
    int4 p4 = *(const int4*)(lpid + b * L_ + l);
    int4 s4 = *(const int4*)(lseg + b * L_ + l);
    int pv[4] = {p4.x, p4.y, p4.z, p4.w};
    int sv[4] = {s4.x, s4.y, s4.z, s4.w};
    v4f mv, rv;
#pragma unroll
    for (int j = 0; j < 4; ++j) {
        int lt  = sv[j] > 0;
        int pid = pv[j];
        mv[j] = (float)((gt == lt) && ((pid == g) || (g == 0)));
        rv[j] = (float)(VOCAB_ + (pid == g ? 1 : 0));
    }
    __builtin_nontemporal_store(mv, (v4f*)(out + OFF_G2L_MASK + f));
    __builtin_nontemporal_store(rv, (v4f*)(out + OFF_G2L_REL  + f));
}

// ------------------------------------------------------------------
extern "C" void kernel_launch(void* const* d_in, const int* in_sizes, int n_in,
                              void* d_out, int out_size, void* d_ws, size_t ws_size,
                              hipStream_t stream) {
    (void)in_sizes; (void)n_in; (void)out_size; (void)ws_size;
    const int* lbp  = (const int*)d_in[0];   // long_paragraph_breakpoints (B,L)
    const int* lpid = (const int*)d_in[1];   // long_paragraph_ids         (B,L)
    const int* gbp  = (const int*)d_in[2];   // global_paragraph_breakpoints (B,G)
    float* out = (float*)d_out;

    int* lseg = (int*)d_ws;                  // B*L ints = 256 KB
    int* gseg = lseg + B_ * L_;              // B*G ints =  16 KB

    seg_kernel<<<B_, 1024, 0, stream>>>(lbp, gbp, lseg, gseg);
    l2l_kernel<<<N_L2L / 4 / 256, 256, 0, stream>>>(lseg, out);
    g2g_kernel<<<(B_ * G_ / 16 * G_ / 16) / 8, 256, 0, stream>>>(gseg, out);
    l2g_kernel<<<N_L2G / 4 / 256, 256, 0, stream>>>(lseg, gseg, lpid, out);
    g2l_kernel<<<N_G2L / 4 / 256, 256, 0, stream>>>(lseg, gseg, lpid, out);
}